// GNNModelV2_79912161509826
// MI455X (gfx1250) — compile-verified
//
#include <hip/hip_runtime.h>
#include <math.h>

#define NNODES 50000
#define NEDGES 500000
#define INCH   64
#define HID    128
#define EPSN   1e-5f

typedef __attribute__((ext_vector_type(2))) float v2f;
typedef __attribute__((ext_vector_type(8))) float v8f;

// ---------------------------------------------------------------- utilities
__global__ void fill_zero_k(float* __restrict__ p, long long n) {
  long long i = blockIdx.x * (long long)blockDim.x + threadIdx.x;
  long long stride = (long long)gridDim.x * blockDim.x;
  for (; i < n; i += stride) p[i] = 0.0f;
}

__global__ void sanitize_k(const float* __restrict__ x, float* __restrict__ y, long long n) {
  long long i = blockIdx.x * (long long)blockDim.x + threadIdx.x;
  if (i >= n) return;
  float v = x[i];
  if (isnan(v)) v = 0.0f;
  else if (isinf(v)) v = copysignf(3.4028234663852886e38f, v);
  y[i] = v;
}

// ------------------------------------------------------------- graph kernels
__global__ void count_deg_k(const long long* __restrict__ ei, float* __restrict__ cnt, int E) {
  int e = blockIdx.x * blockDim.x + threadIdx.x;
  if (e >= E) return;
  long long d = ei[(long long)E + e];
  atomicAdd(&cnt[d], 1.0f);
}

// one thread per (edge, 4-channel group); accumulators live in L2 (25.6MB << 192MB)
__global__ void aggregate_k(const float* __restrict__ h, const long long* __restrict__ ei,
                            float* __restrict__ acc, int E, int C) {
  const int groups = C >> 2;
  long long idx = blockIdx.x * (long long)blockDim.x + threadIdx.x;
  long long total = (long long)E * groups;
  if (idx >= total) return;
  int e = (int)(idx / groups);
  int g = (int)(idx % groups);
  long long s = ei[e];
  long long d = ei[(long long)E + e];
  const float4 v = *(const float4*)(h + s * C + (g << 2));
  float* p = acc + d * C + (g << 2);
  atomicAdd(p + 0, v.x);
  atomicAdd(p + 1, v.y);
  atomicAdd(p + 2, v.z);
  atomicAdd(p + 3, v.w);
}

__global__ void divide_k(float* __restrict__ acc, const float* __restrict__ cnt,
                         long long n, int C) {
  long long i = blockIdx.x * (long long)blockDim.x + threadIdx.x;
  if (i >= n) return;
  float c = cnt[i / C];
  acc[i] = acc[i] / fmaxf(c, 1.0f);
}

// ------------------------------------------------------ fused SAGE WMMA GEMM
// out[M,N] = A[M,K] @ Wa[K,N] + B[M,K] @ Wb[K,N] + bias[N]
// One wave per 16x16 output tile; K stepped by 4 via V_WMMA_F32_16X16X4_F32.
// M % 16 == 0 and N % 16 == 0 guaranteed (50000, 128) -> EXEC all ones, no tails.
__global__ void __launch_bounds__(128)
sage_gemm_wmma_k(const float* __restrict__ A, const float* __restrict__ Wa,
                 const float* __restrict__ B, const float* __restrict__ Wb,
                 const float* __restrict__ bias, float* __restrict__ out,
                 int M, int K, int N) {
  const int lane = threadIdx.x & 31;
  const int wid  = blockIdx.x * (blockDim.x >> 5) + (threadIdx.x >> 5);
  const int tilesN = N >> 4;
  const int tm = wid / tilesN;
  const int tn = wid % tilesN;

  const int r16  = lane & 15;        // A row / B column within tile
  const int kh   = (lane >> 4) << 1; // lanes 0-15 -> K{0,1}, lanes 16-31 -> K{2,3}
  const int ncol = tn * 16 + r16;

  const float* ap = A + (long long)(tm * 16 + r16) * K;
  const float* bp = B + (long long)(tm * 16 + r16) * K;

  v8f acc = {0.f, 0.f, 0.f, 0.f, 0.f, 0.f, 0.f, 0.f};

  for (int k = 0; k < K; k += 4) {
    float2 av = *(const float2*)(ap + k + kh);
    v2f a = {av.x, av.y};
    v2f b = {Wa[(long long)(k + kh) * N + ncol],
             Wa[(long long)(k + kh + 1) * N + ncol]};
    acc = __builtin_amdgcn_wmma_f32_16x16x4_f32(false, a, false, b,
                                                (short)0, acc, false, false);
  }
  for (int k = 0; k < K; k += 4) {
    float2 av = *(const float2*)(bp + k + kh);
    v2f a = {av.x, av.y};
    v2f b = {Wb[(long long)(k + kh) * N + ncol],
             Wb[(long long)(k + kh + 1) * N + ncol]};
    acc = __builtin_amdgcn_wmma_f32_16x16x4_f32(false, a, false, b,
                                                (short)0, acc, false, false);
  }

  const float bv = bias[ncol];
  const int rbase = tm * 16 + ((lane >> 4) << 3); // upper-half lanes hold rows +8
  for (int v = 0; v < 8; ++v)
    out[(long long)(rbase + v) * N + ncol] = acc[v] + bv;
}

// ----------------------------------------------------------------- GraphNorm
// thread-per-column partial sums (coalesced), one atomic per block per column
__global__ void colstats_k(const float* __restrict__ h, float* __restrict__ cs,
                           float* __restrict__ cs2, int M, int C) {
  int c = threadIdx.x; // blockDim.x == C
  float s = 0.f, s2 = 0.f;
  for (int r = blockIdx.x; r < M; r += gridDim.x) {
    float v = h[(long long)r * C + c];
    s += v;
    s2 += v * v;
  }
  atomicAdd(&cs[c], s);
  atomicAdd(&cs2[c], s2);
}

// var = E[(h - a*mu)^2] = E[h^2] - (2a - a^2) * mu^2   (exact, since mu = E[h])
__global__ void finalize_stats_k(const float* __restrict__ cs, const float* __restrict__ cs2,
                                 const float* __restrict__ a, float* __restrict__ mu,
                                 float* __restrict__ inv, int M, int C) {
  int c = threadIdx.x;
  if (c >= C) return;
  float m  = cs[c] / (float)M;
  float e2 = cs2[c] / (float)M;
  float ac = a[c];
  float var = e2 - (2.0f * ac - ac * ac) * m * m;
  mu[c] = m;
  inv[c] = rsqrtf(var + EPSN);
}

__global__ void norm_relu_k(float* __restrict__ h, const float* __restrict__ g,
                            const float* __restrict__ be, const float* __restrict__ a,
                            const float* __restrict__ mu, const float* __restrict__ inv,
                            long long n, int C) {
  long long i = blockIdx.x * (long long)blockDim.x + threadIdx.x;
  if (i >= n) return;
  int c = (int)(i % C);
  float v = (h[i] - a[c] * mu[c]) * inv[c] * g[c] + be[c];
  h[i] = fmaxf(v, 0.0f);
}

// ------------------------------------------------------------------- head
// wave-per-node dot(h[n,:128], Wfc) via float4 + shfl reduction (wave32)
__global__ void head_k(const float* __restrict__ h, const float* __restrict__ Wfc,
                       const float* __restrict__ bfc, float* __restrict__ out, int M) {
  int node = blockIdx.x * (blockDim.x >> 5) + (threadIdx.x >> 5);
  int lane = threadIdx.x & 31;
  if (node >= M) return;
  const float4 v = *(const float4*)(h + (long long)node * HID + (lane << 2));
  const float4 w = *(const float4*)(Wfc + (lane << 2));
  float s = v.x * w.x + v.y * w.y + v.z * w.z + v.w * w.w;
  for (int off = 16; off > 0; off >>= 1) s += __shfl_down(s, off);
  if (lane == 0) out[node] = s + bfc[0];
}

// ------------------------------------------------------------------- driver
extern "C" void kernel_launch(void* const* d_in, const int* in_sizes, int n_in,
                              void* d_out, int out_size, void* d_ws, size_t ws_size,
                              hipStream_t stream) {
  (void)in_sizes; (void)n_in; (void)out_size; (void)ws_size;
  const float*     x   = (const float*)d_in[0];
  const long long* ei  = (const long long*)d_in[1];
  const float *Wl0 = (const float*)d_in[2],  *Wr0 = (const float*)d_in[3];
  const float *b0  = (const float*)d_in[4],  *g0  = (const float*)d_in[5];
  const float *be0 = (const float*)d_in[6],  *a0  = (const float*)d_in[7];
  const float *Wl1 = (const float*)d_in[8],  *Wr1 = (const float*)d_in[9];
  const float *b1  = (const float*)d_in[10], *g1  = (const float*)d_in[11];
  const float *be1 = (const float*)d_in[12], *a1  = (const float*)d_in[13];
  const float *Wfc = (const float*)d_in[14], *bfc = (const float*)d_in[15];
  float* out = (float*)d_out;

  float* ws    = (float*)d_ws;
  float* cnt   = ws;             // NNODES
  float* cs    = ws + 50048;     // 128
  float* cs2   = cs + 128;       // 128
  float* mu    = cs2 + 128;      // 128
  float* inv   = mu + 128;       // 128
  float* R1    = ws + 65536;     // 6.4M floats
  float* R2    = R1 + (long long)NNODES * HID;
  float* R3    = R2 + (long long)NNODES * HID;
  float* h0    = R1;                              // NNODES x 64
  float* mean0 = R1 + (long long)NNODES * INCH;   // NNODES x 64 (accumulator -> mean)
  // R2 = pre0 / h1 (in place), R3 = mean1, R1 reused = pre1 / h2 (h0/mean0 dead)

  const long long nX  = (long long)NNODES * INCH; // 3.2M
  const long long nH  = (long long)NNODES * HID;  // 6.4M

  // ---- layer 0 ----
  fill_zero_k<<<256, 256, 0, stream>>>(cnt, NNODES);
  fill_zero_k<<<1024, 256, 0, stream>>>(mean0, nX);
  sanitize_k<<<(unsigned)((nX + 255) / 256), 256, 0, stream>>>(x, h0, nX);
  count_deg_k<<<(NEDGES + 255) / 256, 256, 0, stream>>>(ei, cnt, NEDGES);
  {
    long long work = (long long)NEDGES * (INCH / 4);
    aggregate_k<<<(unsigned)((work + 255) / 256), 256, 0, stream>>>(h0, ei, mean0, NEDGES, INCH);
  }
  divide_k<<<(unsigned)((nX + 255) / 256), 256, 0, stream>>>(mean0, cnt, nX, INCH);
  {
    int waves = (NNODES / 16) * (HID / 16); // 25000, 4 waves/block
    sage_gemm_wmma_k<<<waves / 4, 128, 0, stream>>>(mean0, Wl0, h0, Wr0, b0, R2,
                                                    NNODES, INCH, HID);
  }
  fill_zero_k<<<1, 256, 0, stream>>>(cs, 256); // cs + cs2 contiguous
  colstats_k<<<240, HID, 0, stream>>>(R2, cs, cs2, NNODES, HID);
  finalize_stats_k<<<1, HID, 0, stream>>>(cs, cs2, a0, mu, inv, NNODES, HID);
  norm_relu_k<<<(unsigned)((nH + 255) / 256), 256, 0, stream>>>(R2, g0, be0, a0, mu, inv, nH, HID);

  // ---- layer 1 ----
  fill_zero_k<<<2048, 256, 0, stream>>>(R3, nH);
  {
    long long work = (long long)NEDGES * (HID / 4);
    aggregate_k<<<(unsigned)((work + 255) / 256), 256, 0, stream>>>(R2, ei, R3, NEDGES, HID);
  }
  divide_k<<<(unsigned)((nH + 255) / 256), 256, 0, stream>>>(R3, cnt, nH, HID);
  {
    int waves = (NNODES / 16) * (HID / 16);
    sage_gemm_wmma_k<<<waves / 4, 128, 0, stream>>>(R3, Wl1, R2, Wr1, b1, R1,
                                                    NNODES, HID, HID);
  }
  fill_zero_k<<<1, 256, 0, stream>>>(cs, 256);
  colstats_k<<<240, HID, 0, stream>>>(R1, cs, cs2, NNODES, HID);
  finalize_stats_k<<<1, HID, 0, stream>>>(cs, cs2, a1, mu, inv, NNODES, HID);
  norm_relu_k<<<(unsigned)((nH + 255) / 256), 256, 0, stream>>>(R1, g1, be1, a1, mu, inv, nH, HID);

  // ---- head ----
  head_k<<<(NNODES + 7) / 8, 256, 0, stream>>>(R1, Wfc, bfc, out, NNODES);
}